// C_iht_28046136442969
// MI455X (gfx1250) — compile-verified
//
#include <hip/hip_runtime.h>
#include <math.h>

#ifndef M_PI
#define M_PI 3.14159265358979323846
#endif

#define N_B    8
#define C_B    128
#define NCTOT  (N_B * C_B)      // 1024
#define A_NUM  180
#define R_NUM  180
#define H_DIM  160
#define W_DIM  160
#define HW     (H_DIM * W_DIM)  // 25600
#define AR     (A_NUM * R_NUM)  // 32400

typedef __attribute__((ext_vector_type(2))) float v2f;
typedef __attribute__((ext_vector_type(8))) float v8f;

// Issue one async B32 copy (memory -> LDS); the 24-bit inst offset advances
// BOTH the LDS and global addresses per the CDNA5 async pseudocode.
#define STAGE_ASYNC(lds, voff, base, OFF)                                   \
  asm volatile("global_load_async_to_lds_b32 %0, %1, %2 offset:" #OFF       \
               :: "v"(lds), "v"(voff), "s"(base) : "memory")

__device__ __forceinline__ void stage9(unsigned lds, unsigned voff,
                                       const float* base) {
  STAGE_ASYNC(lds, voff, base, 0);
  STAGE_ASYNC(lds, voff, base, 80);
  STAGE_ASYNC(lds, voff, base, 160);
  STAGE_ASYNC(lds, voff, base, 240);
  STAGE_ASYNC(lds, voff, base, 320);
  STAGE_ASYNC(lds, voff, base, 400);
  STAGE_ASYNC(lds, voff, base, 480);
  STAGE_ASYNC(lds, voff, base, 560);
  STAGE_ASYNC(lds, voff, base, 640);
}

// One block: 16-nc slab x 4 y-rows x all 160 x. 10 waves; wave w owns the
// 16-pixel strip at x0 = 16*w, holds 4 fp32 accumulators (one per y row).
// Per angle: the 16x180 rho-slice for the NEXT angle is staged into the spare
// LDS buffer with GLOBAL_LOAD_ASYNC_TO_LDS (ASYNCcnt) while the current
// buffer feeds 4 y-rows x 4 K-chunks of v_wmma_f32_16x16x4_f32 with a
// one-hot fp32 selector B built in registers.
__global__ __launch_bounds__(320)
void iht_wmma_kernel(const float* __restrict__ hough, float* __restrict__ out) {
  const int nc0  = blockIdx.y * 16;   // 64 nc blocks
  const int y0   = blockIdx.x * 4;    // 40 y bands
  const int tid  = threadIdx.x;
  const int lane = tid & 31;
  const int wave = tid >> 5;          // 0..9
  const int x0   = wave * 16;
  const int m    = lane & 15;         // A-matrix row M / B-matrix column N
  const int koff = (lane < 16) ? 0 : 2;  // K slot held in V0 per ISA layout

  __shared__ float s_tc[A_NUM];
  __shared__ float s_ts[A_NUM];
  __shared__ float tile[2][16 * R_NUM];  // double buffer, 2 x 11520 B

  // Per-angle constants in f64, matching the reference's numpy f64 table.
  if (tid < A_NUM) {
    double th       = (double)tid * (M_PI / (double)A_NUM);
    double inv_irho = (double)R_NUM / 227.0;  // irho = (int(sqrt(2*160^2))+1)/180
    s_tc[tid] = (float)(cos(th) * inv_irho);
    s_ts[tid] = (float)(sin(th) * inv_irho);
  }

  // Staging map: 320 threads = 16 rows x 20 cols, 9 strided copies each.
  const int frow = tid / 20;   // 0..15
  const int fcol = tid % 20;   // 0..19
  const unsigned ldsA =
      (unsigned)(uintptr_t)(void*)&tile[0][frow * R_NUM + fcol];
  const unsigned ldsB =
      (unsigned)(uintptr_t)(void*)&tile[1][frow * R_NUM + fcol];
  const unsigned voff0 =
      ((unsigned)(nc0 + frow) * (unsigned)AR + (unsigned)fcol) * 4u;

  const float xc = (float)(x0 + m - W_DIM / 2);

  v8f accv[4] = {};

  // Kick off the stage for angle 0 into buffer 0.
  stage9(ldsA, voff0, hough);

  __syncthreads();  // cos/sin tables ready

  for (int a = 0; a < A_NUM; ++a) {
    const int buf = a & 1;
    if (a + 1 < A_NUM) {
      // Stage next angle into the spare buffer, then retire the current
      // stage: async loads complete in order, so <=9 outstanding means the
      // 9 ops of stage(a) are done.
      stage9(buf ? ldsA : ldsB, voff0 + (unsigned)(a + 1) * 720u, hough);
      asm volatile("s_wait_asynccnt 0x9" ::: "memory");
    } else {
      asm volatile("s_wait_asynccnt 0x0" ::: "memory");
    }
    __syncthreads();

    const float tc = s_tc[a];
    const float ts = s_ts[a];

#pragma unroll
    for (int yy = 0; yy < 4; ++yy) {
      const float ycf = (float)(y0 + yy - H_DIM / 2);
      const float raw = fmaf(ycf, ts, xc * tc);
      const int   ri  = (int)rintf(raw) + R_NUM / 2;  // RNE == np.round

      // r is affine in x -> strip min is at an endpoint. A 16-bin window
      // always covers all 16 pixels (max spread <= 13) and stays in [0,180).
      int r0 = min(__builtin_amdgcn_readlane(ri, 0),
                   __builtin_amdgcn_readlane(ri, 15));
      r0 = min(max(r0, 0), R_NUM - 16);
      const int rr = min(max(ri, 0), R_NUM - 1) - r0;  // 0..15, one-hot slot

      const float* arow = &tile[buf][m * R_NUM + r0];
      v8f acc = accv[yy];
#pragma unroll
      for (int c4 = 0; c4 < 4; ++c4) {
        const int k0 = 4 * c4 + koff;
        v2f Af = { arow[k0], arow[k0 + 1] };                 // A: H window
        v2f Bf = { (rr == k0)     ? 1.0f : 0.0f,             // B: one-hot sel
                   (rr == k0 + 1) ? 1.0f : 0.0f };
        acc = __builtin_amdgcn_wmma_f32_16x16x4_f32(
                  false, Af, false, Bf, (short)0, acc, false, false);
      }
      accv[yy] = acc;
    }
    __syncthreads();  // all reads of tile[buf] done before it is restaged
  }

  // C/D layout: VGPR v holds M = v (lanes 0-15) or v+8 (lanes 16-31), N = m.
  const int mbase = (lane < 16) ? 0 : 8;
#pragma unroll
  for (int yy = 0; yy < 4; ++yy) {
    const size_t ybase = (size_t)(y0 + yy) * W_DIM + (x0 + m);
#pragma unroll
    for (int v = 0; v < 8; ++v) {
      const int M = v + mbase;
      out[(size_t)(nc0 + M) * HW + ybase] = accv[yy][v];
    }
  }
}

extern "C" void kernel_launch(void* const* d_in, const int* in_sizes, int n_in,
                              void* d_out, int out_size, void* d_ws, size_t ws_size,
                              hipStream_t stream) {
  (void)in_sizes; (void)n_in; (void)d_ws; (void)ws_size; (void)out_size;
  const float* hough = (const float*)d_in[0];
  float* out = (float*)d_out;
  dim3 grid(H_DIM / 4, NCTOT / 16);  // (40 y-bands, 64 nc blocks)
  iht_wmma_kernel<<<grid, 320, 0, stream>>>(hough, out);
}